// SiddonProjector3D_24215025615245
// MI455X (gfx1250) — compile-verified
//
#include <hip/hip_runtime.h>

// Siddon 3D forward projector for MI455X (gfx1250, wave32).
// One ray per wave32. Per 4 traversal steps:
//   - wave-uniform Siddon stepping, branch-free (signs/offsets/step/scale and
//     reciprocals hoisted; per-step crossing distance = one FMA + multiply)
//   - 8 coalesced b32 gathers of the contiguous 64-float depth line per cell
//   - 4x V_WMMA_F32_16X16X4_F32:  D(16x16) += A(16x4 = step weights, rows
//     replicated) x B(4x16 = gathered volume, one 16-wide v-chunk per call)
// The early-exit condition is broadcast via readfirstlane so the loop branch
// is scalar (s_cbranch, no exec-mask churn) and EXEC is structurally all-1s
// around every WMMA.

typedef __attribute__((ext_vector_type(2))) float v2f;
typedef __attribute__((ext_vector_type(8))) float v8f;

#define A_N 90
#define U_N 96
#define V_N 64
#define W_N 96
#define H_N 96
#define D_N 64
#define NSTEPS 194
#define NBATCH 49          // 49*4 = 196 >= NSTEPS; padded steps have w == 0
#define EPSF 1e-12f
#define RAYS_PER_BLOCK 8   // 8 wave32 per 256-thread block

__global__ __launch_bounds__(256)
void siddon_fp_wmma_kernel(const float* __restrict__ vol, float* __restrict__ out)
{
    const int lane = threadIdx.x & 31;
    const int wave = threadIdx.x >> 5;
    const int ray  = blockIdx.x * RAYS_PER_BLOCK + wave;   // 1080*8 == 8640 exactly
    const int a    = ray / U_N;
    const int uidx = ray - a * U_N;

    // ---- ray setup (matches _trace()) ----
    const float angle = (float)a * (3.14159265358979323846f / (float)A_N);
    const float dxr = cosf(angle);
    const float dyr = sinf(angle);
    const float u   = (float)uidx - (float)(U_N - 1) * 0.5f;   // SU = 1
    const float x0  = -u * dyr;
    const float y0  =  u * dxr;

    const float cxw = (float)(W_N - 1) * 0.5f;   // SX = SY = 1
    const float cyh = (float)(H_N - 1) * 0.5f;
    const float x_min = -cxw, x_max = cxw;
    const float y_min = -cyh, y_max = cyh;
    const float DIAG = 1.41421356237309504880f;
    const float INF  = __builtin_inff();

    // _t_for for x axis
    float tx0, tx1;
    {
        bool  par  = fabsf(dxr) < EPSF;
        float safe = par ? 1.0f : dxr;
        float t0 = (x_min - x0) / safe, t1 = (x_max - x0) / safe;
        float lo = fminf(t0, t1), hi = fmaxf(t0, t1);
        bool inside = (x0 >= x_min) && (x0 <= x_max);
        tx0 = par ? (inside ? -INF : INF) : lo;
        tx1 = par ? (inside ?  INF : -INF) : hi;
    }
    // _t_for for y axis
    float ty0, ty1;
    {
        bool  par  = fabsf(dyr) < EPSF;
        float safe = par ? 1.0f : dyr;
        float t0 = (y_min - y0) / safe, t1 = (y_max - y0) / safe;
        float lo = fminf(t0, t1), hi = fmaxf(t0, t1);
        bool inside = (y0 >= y_min) && (y0 <= y_max);
        ty0 = par ? (inside ? -INF : INF) : lo;
        ty1 = par ? (inside ?  INF : -INF) : hi;
    }

    const float t_entry = fmaxf(tx0, ty0);
    const float t_exit  = fminf(tx1, ty1);
    bool  alive = t_entry < t_exit;
    const float te  = alive ? t_entry : 0.0f;
    const float tex = alive ? t_exit  : 0.0f;
    const float texEps = tex - EPSF;

    float x = x0 + te * dxr;
    float y = y0 + te * dyr;
    float t = te;
    int ii = (int)fminf(fmaxf(rintf(x + cxw), 0.0f), (float)(W_N - 1));
    int jj = (int)fminf(fmaxf(rintf(y + cyh), 0.0f), (float)(H_N - 1));

    // ---- loop-invariant traversal constants ----
    const bool  okx = fabsf(dxr) > EPSF;
    const bool  oky = fabsf(dyr) > EPSF;
    const float rdx = 1.0f / dxr;                 // used only when okx
    const float rdy = 1.0f / dyr;                 // used only when oky
    const float xoff = ((dxr > 0.0f) ? 0.5f : -0.5f) - cxw;  // xn = fi + xoff
    const float yoff = ((dyr > 0.0f) ? 0.5f : -0.5f) - cyh;
    const int   sx = (dxr > 0.0f) ? 1 : -1;
    const int   sy = (dyr > 0.0f) ? 1 : -1;
    const float segScale = DIAG / fmaxf(fabsf(dxr) + fabsf(dyr), EPSF);

    // ---- WMMA accumulators: 4 chunks of 16 v-values (all 16 rows identical) ----
    v8f zero8 = 0.0f;
    v8f acc[4];
    acc[0] = zero8; acc[1] = zero8; acc[2] = zero8; acc[3] = zero8;

    const bool half16 = (lane >= 16);   // lanes 16..31 own K = {2,3}
    const int  ncol   = lane & 15;

    for (int it = 0; it < NBATCH; ++it) {
        int   bi[4], bj[4];
        float bw[4];
        // 4 Siddon steps (wave-uniform, branch-free)
        #pragma unroll
        for (int k = 0; k < 4; ++k) {
            bool  valid = alive && (t < texEps);
            float fi = (float)ii, fj = (float)jj;
            float tx = okx ? (fi + xoff - x) * rdx : INF;
            float ty = oky ? (fj + yoff - y) * rdy : INF;
            float dt = fminf(fminf(tx, ty), tex - t);
            float seg = fmaxf(0.0f, dt * segScale);
            bi[k] = ii;
            bj[k] = jj;
            bw[k] = valid ? seg : 0.0f;
            int i_n = ii + ((tx <= ty) ? sx : 0);
            int j_n = jj + ((ty <= tx) ? sy : 0);
            bool inb = ((unsigned)i_n < (unsigned)W_N) && ((unsigned)j_n < (unsigned)H_N);
            float dts = valid ? dt : 0.0f;
            ii = valid ? i_n : ii;
            jj = valid ? j_n : jj;
            x += dxr * dts;
            y += dyr * dts;
            t += dts;
            alive = valid && inb;
        }

        // Per-lane K selection: lanes 0-15 -> steps {0,1}, lanes 16-31 -> {2,3}
        int ilo = half16 ? bi[2] : bi[0];
        int jlo = half16 ? bj[2] : bj[0];
        int ihi = half16 ? bi[3] : bi[1];
        int jhi = half16 ? bj[3] : bj[1];
        // Clamp for the load address only (weight is already 0 for OOB steps)
        ilo = min(max(ilo, 0), W_N - 1);  jlo = min(max(jlo, 0), H_N - 1);
        ihi = min(max(ihi, 0), W_N - 1);  jhi = min(max(jhi, 0), H_N - 1);
        const int blo = ilo * (H_N * D_N) + jlo * D_N;
        const int bhi = ihi * (H_N * D_N) + jhi * D_N;

        v2f av;
        av.x = half16 ? bw[2] : bw[0];
        av.y = half16 ? bw[3] : bw[1];

        #pragma unroll
        for (int c = 0; c < 4; ++c) {
            v2f bv;
            bv.x = vol[blo + c * 16 + ncol];
            bv.y = vol[bhi + c * 16 + ncol];
            // 8 args: (neg_a, A, neg_b, B, c_mod, C, reuse_a, reuse_b)
            acc[c] = __builtin_amdgcn_wmma_f32_16x16x4_f32(
                false, av, false, bv, (short)0, acc[c], false, false);
        }

        // 'alive' is wave-uniform by construction; broadcast it so the loop
        // exit is a scalar branch (no exec-mask bookkeeping, EXEC stays all-1s
        // for the WMMAs above).
        if (!__builtin_amdgcn_readfirstlane((int)alive)) break;
    }

    // All 16 rows of each D are identical; row M=0 lives in element [0],
    // lanes 0-15 holding N = 0..15.  out[(u*A + a)*V + v]
    const int outBase = __builtin_amdgcn_readfirstlane((uidx * A_N + a) * V_N);
    if (lane < 16) {
        float* o = out + outBase;
        #pragma unroll
        for (int c = 0; c < 4; ++c)
            o[c * 16 + lane] = acc[c][0];
    }
}

extern "C" void kernel_launch(void* const* d_in, const int* in_sizes, int n_in,
                              void* d_out, int out_size, void* d_ws, size_t ws_size,
                              hipStream_t stream) {
    const float* vol = (const float*)d_in[0];
    float* out = (float*)d_out;
    dim3 block(256);
    dim3 grid((A_N * U_N) / RAYS_PER_BLOCK);   // 8640 / 8 = 1080 blocks
    siddon_fp_wmma_kernel<<<grid, block, 0, stream>>>(vol, out);
}